// Terrain3D_33483565040111
// MI455X (gfx1250) — compile-verified
//
#include <hip/hip_runtime.h>
#include <hip/hip_bf16.h>

// ---------------- problem constants (from reference) ----------------
#define G        128
#define G3       (G * G * G)
#define NEC      4
#define RAD      9          // RADIUS = int(3*0.05*128/2) = 9
#define KW       19         // W = 2*RADIUS+1
#define ETA_C    0.01f
#define LEAK_C   5e-05f
#define AH_C     0.002f
#define AE_C     0.001f
// sig = SIGMA*G/2 = 3.2 ;  1/(2*sig^2)
#define INV2SIG2 (1.0f / (2.0f * 3.2f * 3.2f))
#define PAD      16         // float pad before/after the 5-grid region

typedef __attribute__((ext_vector_type(2))) float v2f;
typedef __attribute__((ext_vector_type(8))) float v8f;

// ---------------- init: load H0 / E0 into workspace grids ----------------
__global__ void terrain_init_kernel(const float* __restrict__ H0,
                                    const float* __restrict__ E0,
                                    float* __restrict__ Hg,
                                    float* __restrict__ Eg) {
    size_t i = (size_t)blockIdx.x * blockDim.x + threadIdx.x;
    size_t stride = (size_t)gridDim.x * blockDim.x;
    for (size_t k = i; k < (size_t)G3; k += stride)            Hg[k] = H0[k];
    for (size_t k = i; k < (size_t)NEC * G3; k += stride)      Eg[k] = E0[k];
}

// ---------------- splat: one block per particle ----------------
// 128 threads = 4 wave32. Each wave computes one 16x16 tile of the separable
// xy-plane kx (outer) ky with V_WMMA_F32_16X16X4_F32 (A: kx in column K=0,
// B: ky in row K=0), then walks the contiguous z-column with branchless,
// fully unrolled f32 atomics (invalid z taps have weight 0 -> atomic +0.0 is
// a numerical no-op; a 16-float workspace pad absorbs the <=9-element address
// overhang at the global extremes).
__global__ __launch_bounds__(128) void terrain_splat_kernel(
        const float* __restrict__ pos,
        const float* __restrict__ inten,
        const float* __restrict__ emo,
        float* __restrict__ Hc,
        float* __restrict__ E0c,
        float* __restrict__ E1c,
        float* __restrict__ E2c,
        float* __restrict__ E3c) {
    __shared__ float kw_s[3][KW];   // Gaussian weights (0 if coord invalid)
    __shared__ int   kc_s[3][KW];   // grid coordinates (may be out of range)
    __shared__ float s_omega;
    __shared__ float s_emo[NEC];

    const int p = blockIdx.x;
    const int t = threadIdx.x;

    if (t == 0)   s_omega = inten[p] * ETA_C;
    if (t < NEC)  s_emo[t] = emo[p * NEC + t];
    if (t < 3 * KW) {
        int dim = t / KW, j = t % KW;
        float gp = (pos[p * 3 + dim] + 1.0f) * 0.5f * (float)(G - 1);
        int c = (int)floorf(gp);
        c = min(max(c, 0), G - 1);
        int coord = c + j - RAD;
        float d = (float)coord - gp;
        float w = 0.0f;
        if (coord >= 0 && coord < G)
            w = expf(-(d * d) * INV2SIG2);
        kc_s[dim][j] = coord;
        kw_s[dim][j] = w;
    }
    __syncthreads();

    const int lane = t & 31;
    const int wave = t >> 5;        // 0..3
    const int tm = wave >> 1;       // row tile of the 19x19 plane
    const int tn = wave & 1;        // col tile
    const int half = lane >> 4;     // 0: lanes 0-15, 1: lanes 16-31
    const int l15 = lane & 15;

    // A (16x4 f32): lanes 0-15 hold K=0 (VGPR0) / K=1 (VGPR1); lanes 16-31 K=2/K=3.
    // Only K=0 is nonzero -> rank-1 product kx (outer) ky.
    v2f a; a.x = 0.0f; a.y = 0.0f;
    if (half == 0) { int m = tm * 16 + l15; if (m < KW) a.x = kw_s[0][m]; }
    // B (4x16 f32): row K=0 striped over lanes 0-15 of VGPR0.
    v2f b; b.x = 0.0f; b.y = 0.0f;
    if (half == 0) { int n = tn * 16 + l15; if (n < KW) b.x = kw_s[1][n]; }

    v8f acc = {};
    // EXEC is all-ones here (full 128-thread block, no divergence at this point)
    acc = __builtin_amdgcn_wmma_f32_16x16x4_f32(
        /*neg_a=*/false, a, /*neg_b=*/false, b,
        /*c_mod=*/(short)0, acc, /*reuse_a=*/false, /*reuse_b=*/false);

    const float omega = s_omega;
    const float e0 = s_emo[0], e1 = s_emo[1], e2 = s_emo[2], e3 = s_emo[3];

    // Uniform z-column start: gz(z) = cz - RAD + z; weights are 0 for gz
    // outside [0,G). idx0 may under/overhang the grid by <= RAD elements,
    // covered by the workspace pad (those taps add exactly +0.0).
    const int cz = kc_s[2][RAD];          // clipped center z coordinate
    const int zbase = cz - RAD;

    // C/D layout: VGPR r, lanes 0-15 -> M=r, N=lane ; lanes 16-31 -> M=r+8, N=lane-16
    #pragma unroll
    for (int r = 0; r < 8; ++r) {
        int m = tm * 16 + r + 8 * half;
        int n = tn * 16 + l15;
        if (m >= KW || n >= KW) continue;
        int gx = kc_s[0][m];
        int gy = kc_s[1][n];
        if (gx < 0 || gx >= G || gy < 0 || gy >= G) continue;
        float wxy = acc[r] * omega;           // omega * kx[m] * ky[n]
        float w0 = wxy * e0, w1 = wxy * e1, w2 = wxy * e2, w3 = wxy * e3;
        int idx0 = (gx * G + gy) * G + zbase; // shared 32-bit voffset; z*4 folds into IOFFSET
        #pragma unroll
        for (int z = 0; z < KW; ++z) {
            float kzv = kw_s[2][z];
            atomicAdd(&Hc [idx0 + z], wxy * kzv);
            atomicAdd(&E0c[idx0 + z], w0 * kzv);
            atomicAdd(&E1c[idx0 + z], w1 * kzv);
            atomicAdd(&E2c[idx0 + z], w2 * kzv);
            atomicAdd(&E3c[idx0 + z], w3 * kzv);
        }
    }
}

// ---------------- fused step + trilinear sample ----------------
__device__ __forceinline__ int clampg(int v) {
    return v < 0 ? 0 : (v > G - 1 ? G - 1 : v);
}
__device__ __forceinline__ float vol_at(const float* __restrict__ v,
                                        int a, int b, int c) {
    return v[(clampg(a) * G + clampg(b)) * G + clampg(c)];
}
// stepped(vol)[a,b,c] = max(vol*(1-LEAK) + bias + alpha*lap_edge(vol), 0)
__device__ __forceinline__ float stepped_at(const float* __restrict__ v,
                                            int a, int b, int c,
                                            float alpha, float bias) {
    float ctr = v[(a * G + b) * G + c];
    float lap = vol_at(v, a - 1, b, c) + vol_at(v, a + 1, b, c)
              + vol_at(v, a, b - 1, c) + vol_at(v, a, b + 1, c)
              + vol_at(v, a, b, c - 1) + vol_at(v, a, b, c + 1)
              - 6.0f * ctr;
    return fmaxf(ctr * (1.0f - LEAK_C) + bias + alpha * lap, 0.0f);
}

__global__ void terrain_sample_kernel(const float* __restrict__ sp,
                                      const float* __restrict__ Hg,
                                      const float* __restrict__ Eg,
                                      float* __restrict__ out, int S) {
    int s = blockIdx.x * blockDim.x + threadIdx.x;
    if (s >= S) return;

    int i0[3], i1[3];
    float fr[3];
    #pragma unroll
    for (int d = 0; d < 3; ++d) {
        float tv = (sp[s * 3 + d] + 1.0f) * 0.5f * (float)(G - 1);
        tv = fminf(fmaxf(tv, 0.0f), (float)(G - 1));
        int ii = (int)floorf(tv);
        fr[d] = tv - (float)ii;
        ii = clampg(ii);
        i0[d] = ii;
        i1[d] = min(ii + 1, G - 1);
    }

    // NOTE: reference indexes vol[..., zi, yi, xi]:
    // slowest axis <- z-component of the sample point, fastest <- x-component.
    float accv[1 + NEC] = {0.0f, 0.0f, 0.0f, 0.0f, 0.0f};
    #pragma unroll
    for (int dz = 0; dz < 2; ++dz) {
        int a = dz ? i1[2] : i0[2];
        float wz = dz ? fr[2] : 1.0f - fr[2];
        #pragma unroll
        for (int dy = 0; dy < 2; ++dy) {
            int b = dy ? i1[1] : i0[1];
            float wy = dy ? fr[1] : 1.0f - fr[1];
            #pragma unroll
            for (int dx = 0; dx < 2; ++dx) {
                int c = dx ? i1[0] : i0[0];
                float wx = dx ? fr[0] : 1.0f - fr[0];
                float w = wz * wy * wx;
                accv[0] += stepped_at(Hg, a, b, c, AH_C, 0.0f) * w;
                #pragma unroll
                for (int e = 0; e < NEC; ++e)
                    accv[1 + e] += stepped_at(Eg + (size_t)e * G3, a, b, c,
                                              AE_C, LEAK_C) * w;
            }
        }
    }
    #pragma unroll
    for (int k = 0; k < 1 + NEC; ++k)
        out[(size_t)s * (1 + NEC) + k] = accv[k];
}

// ---------------- launch ----------------
extern "C" void kernel_launch(void* const* d_in, const int* in_sizes, int n_in,
                              void* d_out, int out_size, void* d_ws, size_t ws_size,
                              hipStream_t stream) {
    const float* positions   = (const float*)d_in[0];
    const float* intensities = (const float*)d_in[1];
    const float* emotions    = (const float*)d_in[2];
    const float* samplepos   = (const float*)d_in[3];
    const float* H0          = (const float*)d_in[4];
    const float* E0          = (const float*)d_in[5];

    // Layout: [pad 16f][ H: G^3 ][ E0..E3: 4*G^3 ][pad 16f]
    float* Hg = (float*)d_ws + PAD;
    float* Eg = Hg + (size_t)G3;

    int N = in_sizes[1];                 // number of particles
    int S = in_sizes[3] / 3;             // number of sample points

    terrain_init_kernel<<<2048, 256, 0, stream>>>(H0, E0, Hg, Eg);
    terrain_splat_kernel<<<N, 128, 0, stream>>>(
        positions, intensities, emotions,
        Hg,
        Eg + 0 * (size_t)G3, Eg + 1 * (size_t)G3,
        Eg + 2 * (size_t)G3, Eg + 3 * (size_t)G3);
    terrain_sample_kernel<<<(S + 255) / 256, 256, 0, stream>>>(samplepos, Hg,
                                                               Eg,
                                                               (float*)d_out, S);
}